// GRTEmbeddingBag_14688788152582
// MI455X (gfx1250) — compile-verified
//
#include <hip/hip_runtime.h>
#include <hip/hip_bf16.h>

typedef float v2f __attribute__((ext_vector_type(2)));
typedef float v4f __attribute__((ext_vector_type(4)));
typedef float v8f __attribute__((ext_vector_type(8)));

#define EMB_DIMS 128
#define NUM_BAGS 4096
#define TTILE 512   // 16x32 floats per (d0,d1) pair
// T workspace is sized for 2600 "pairs": 2500 real + 100 dump slots that
// absorb the clamped/overshoot rows of the precompute tiles (never read).

// ---------------------------------------------------------------------------
// Kernel A: precompute T[(d0,d1)][k=x*4+y][s] = sum_r g0[d0][x][r]*g1[d1][r][y][s]
// A = core0 viewed as [200 x 32] (row g = d0*4+x is contiguous), B = core1[d1] [32 x 128].
// Grid: 50 d1 * 13 m-tiles * 8 n-tiles, one wave per 16x16 C tile.
// A rows >= 200 load from a clamped address (row 199): garbage pollutes only
// C rows >= 200, which are stored unconditionally into the dump region
// (pair index 2500..2599) -> zero branches in the whole kernel body.
// ---------------------------------------------------------------------------
__global__ void tt_precompute_kernel(const float* __restrict__ core0,
                                     const float* __restrict__ core1,
                                     float* __restrict__ T) {
  const int lane = threadIdx.x & 31;
  const int b    = blockIdx.x;            // 50*104
  const int d1   = b / 104;
  const int rem  = b % 104;
  const int mt   = rem >> 3;              // 0..12  (M = 200 rows)
  const int nt   = rem & 7;               // 0..7   (N = 128 cols)
  const int m    = lane & 15;
  const int hi   = lane >> 4;
  const int col  = nt * 16 + m;           // global N column for B / C
  int grow = mt * 16 + m;                 // global M row for A
  if (grow > 199) grow = 199;             // clamp: garbage rows go to dump slots
  const float* Ap = core0 + grow * 32 + hi * 2;
  const float* Bp = core1 + d1 * 4096 + col;

  v2f a[8], bb[8];
#pragma unroll
  for (int kc = 0; kc < 8; ++kc) {        // batch all loads -> one clause
    a[kc] = *(const v2f*)(Ap + kc * 4);
    const int kk = kc * 4 + hi * 2;
    bb[kc].x = Bp[kk * 128];
    bb[kc].y = Bp[(kk + 1) * 128];
  }
  v8f c = {};
#pragma unroll
  for (int kc = 0; kc < 8; ++kc)
    c = __builtin_amdgcn_wmma_f32_16x16x4_f32(false, a[kc], false, bb[kc],
                                              (short)0, c, false, false);

  // scatter C into [pair][16][32] layout: row g=d0*4+x, col=y*32+s -> T[pair][x*4+y][s]
  // gr in [200,207] => pair in [2500,2599]: lands in the dump region, no guard.
  const int s = col & 31;
  const int y = col >> 5;
#pragma unroll
  for (int j = 0; j < 8; ++j) {
    const int gr = mt * 16 + j + hi * 8;  // C/D: vgpr j -> M=j (lo lanes) / M=j+8 (hi)
    const int d0 = gr >> 2;
    const int x  = gr & 3;
    T[(d0 * 50 + d1) * TTILE + ((x << 2) | y) * 32 + s] = c[j];
  }
}

// ---------------------------------------------------------------------------
// Kernel B: cache-table bags. One wave per bag; lane owns 4 consecutive
// columns (b128 loads). 4-deep unroll with independent accumulators keeps
// 4 gathers in flight per thread (HBM bandwidth is the limiter: ~105 MB).
// Fully initializes d_out (the TT kernel then read-modify-writes it).
// ---------------------------------------------------------------------------
__global__ void cache_bag_kernel(const int* __restrict__ cidx,
                                 const int* __restrict__ coff,
                                 const float* __restrict__ table,
                                 float* __restrict__ out) {
  const int lane = threadIdx.x & 31;
  const int wave = threadIdx.x >> 5;
  const int bag  = blockIdx.x * 8 + wave;
  const int s    = coff[bag];
  const int e    = coff[bag + 1];
  const size_t colb = (size_t)lane * 4;

  v4f a0 = {}, a1 = {}, a2 = {}, a3 = {};
  int i = s;
  for (; i + 3 < e; i += 4) {
    a0 += *(const v4f*)(table + (size_t)cidx[i]     * EMB_DIMS + colb);
    a1 += *(const v4f*)(table + (size_t)cidx[i + 1] * EMB_DIMS + colb);
    a2 += *(const v4f*)(table + (size_t)cidx[i + 2] * EMB_DIMS + colb);
    a3 += *(const v4f*)(table + (size_t)cidx[i + 3] * EMB_DIMS + colb);
  }
  for (; i < e; ++i)
    a0 += *(const v4f*)(table + (size_t)cidx[i] * EMB_DIMS + colb);
  a0 += a1 + a2 + a3;
  *(v4f*)(out + (size_t)bag * EMB_DIMS + colb) = a0;
}

// ---------------------------------------------------------------------------
// Kernel C: TT bags. One wave per bag (256-thread block = 8 bags). Per index:
// emb(16x8) = T[pair](16x32) @ g2[d2](32x8) via 8 chained WMMA_F32_16X16X4;
// the WMMA accumulator carries the bag segment-sum. B columns 8..15 simply
// duplicate columns 0..7 (mcol = m&7): they pollute only C columns we never
// store -> no masking, no divergence around the WMMAs. core2 (51.2 KB) is
// staged in LDS so B operands come from ds_load.
// ---------------------------------------------------------------------------
__global__ void tt_bag_kernel(const int* __restrict__ idx,
                              const int* __restrict__ off,
                              const float* __restrict__ T,
                              const float* __restrict__ core2,
                              float* __restrict__ out) {
  __shared__ float g2s[50 * 256];         // 51.2 KB of 320 KB WGP LDS
  for (int i = threadIdx.x; i < 3200; i += 256)
    ((v4f*)g2s)[i] = ((const v4f*)core2)[i];
  __syncthreads();

  const int lane = threadIdx.x & 31;
  const int wave = threadIdx.x >> 5;
  const int bag  = blockIdx.x * 8 + wave;
  const int s0   = off[bag];
  const int e0   = off[bag + 1];
  const int m    = lane & 15;             // A row / B,C column
  const int hi   = lane >> 4;
  const int mcol = m & 7;                 // clamped B column (dup cols 8..15)

  v8f c0 = {}, c1 = {};
  int i = s0;
  for (; i + 1 < e0; i += 2) {
    const int idA = idx[i], idB = idx[i + 1];
    const float* TtA = T + ((idA / 2500) * 50 + (idA / 50) % 50) * TTILE + m * 32 + hi * 2;
    const float* TtB = T + ((idB / 2500) * 50 + (idB / 50) % 50) * TTILE + m * 32 + hi * 2;
    const float* G2A = g2s + (idA % 50) * 256 + hi * 16 + mcol;
    const float* G2B = g2s + (idB % 50) * 256 + hi * 16 + mcol;
    v2f aA[8], aB[8], bA[8], bB[8];
#pragma unroll
    for (int kc = 0; kc < 8; ++kc) {      // batch all 32 loads first
      aA[kc] = *(const v2f*)(TtA + kc * 4);
      aB[kc] = *(const v2f*)(TtB + kc * 4);
      bA[kc].x = G2A[kc * 32];  bA[kc].y = G2A[kc * 32 + 8];
      bB[kc].x = G2B[kc * 32];  bB[kc].y = G2B[kc * 32 + 8];
    }
#pragma unroll
    for (int kc = 0; kc < 8; ++kc) {      // 16 WMMAs, two independent chains
      c0 = __builtin_amdgcn_wmma_f32_16x16x4_f32(false, aA[kc], false, bA[kc],
                                                 (short)0, c0, false, false);
      c1 = __builtin_amdgcn_wmma_f32_16x16x4_f32(false, aB[kc], false, bB[kc],
                                                 (short)0, c1, false, false);
    }
  }
  if (i < e0) {                           // tail index
    const int id = idx[i];
    const float* Tt = T + ((id / 2500) * 50 + (id / 50) % 50) * TTILE + m * 32 + hi * 2;
    const float* G2 = g2s + (id % 50) * 256 + hi * 16 + mcol;
    v2f a[8], bb[8];
#pragma unroll
    for (int kc = 0; kc < 8; ++kc) {
      a[kc] = *(const v2f*)(Tt + kc * 4);
      bb[kc].x = G2[kc * 32];  bb[kc].y = G2[kc * 32 + 8];
    }
#pragma unroll
    for (int kc = 0; kc < 8; ++kc)
      c0 = __builtin_amdgcn_wmma_f32_16x16x4_f32(false, a[kc], false, bb[kc],
                                                 (short)0, c0, false, false);
  }
  c0 = c0 + c1;
  if (m < 8) {
    // C/D layout: vgpr j -> M=j (lanes 0-15) / M=j+8 (lanes 16-31), N=lane&15.
    float* o = out + (size_t)bag * EMB_DIMS + m;   // emb index e = k*8 + z
#pragma unroll
    for (int j = 0; j < 8; ++j) {
      const int row = j + hi * 8;
      o[row * 8] += c0[j];
    }
  }
}

// ---------------------------------------------------------------------------
extern "C" void kernel_launch(void* const* d_in, const int* in_sizes, int n_in,
                              void* d_out, int out_size, void* d_ws, size_t ws_size,
                              hipStream_t stream) {
  const int*   indices        = (const int*)d_in[0];
  const int*   offsets        = (const int*)d_in[1];
  const int*   cached_indices = (const int*)d_in[2];
  const int*   cached_offsets = (const int*)d_in[3];
  const float* core0          = (const float*)d_in[4];
  const float* core1          = (const float*)d_in[5];
  const float* core2          = (const float*)d_in[6];
  const float* cache_table    = (const float*)d_in[7];
  float* out = (float*)d_out;
  float* T   = (float*)d_ws;              // needs 2600*512*4 = 5.33 MB scratch
                                          // (2500 real pairs + 100 dump slots)

  // 1) precompute T[d0,d1] = g0 @ g1  (50*13*8 waves, fully covers T)
  tt_precompute_kernel<<<50 * 104, 32, 0, stream>>>(core0, core1, T);
  // 2) cache bags: writes (initializes) every element of d_out
  cache_bag_kernel<<<NUM_BAGS / 8, 256, 0, stream>>>(cached_indices, cached_offsets,
                                                     cache_table, out);
  // 3) TT bags: accumulates into d_out (stream order guarantees 1,2 done)
  tt_bag_kernel<<<NUM_BAGS / 8, 256, 0, stream>>>(indices, offsets, T, core2, out);
}